// Swap_18957985644706
// MI455X (gfx1250) — compile-verified
//
#include <hip/hip_runtime.h>

// Shapes from the reference: x is (N=32, C=256, H=64, W=64) float32.
#define N_  32
#define C_  256
#define Q_  64      // channels per group
#define H_  64
#define W_  64
#define W4_ 16      // float4 slots per row

typedef float float4v __attribute__((ext_vector_type(4)));

// -----------------------------------------------------------------------------
// Kernel A: output channels [0,128). Roll along H => whole contiguous rows are
// copied (row = 64 floats = 256B). Pure aligned b128 load -> b128 store, with
// non-temporal hints (data is touched exactly once; don't churn L2).
// One float4 per thread. idx bits: [t:4][h:6][oc:7][n:5] = 22 bits = 4,194,304.
// -----------------------------------------------------------------------------
__global__ void __launch_bounds__(256)
roll_h_kernel(const float* __restrict__ x, float* __restrict__ out) {
    int idx = blockIdx.x * 256 + threadIdx.x;
    int t   = idx & (W4_ - 1);
    int h   = (idx >> 4) & (H_ - 1);
    int oc  = (idx >> 10) & 127;       // output channel 0..127
    int n   = idx >> 17;

    int ic, sh;
    if (oc < Q_) { ic = oc + Q_; sh = (h - 1) & (H_ - 1); }  // y0 = roll(g1,+1,H)
    else         { ic = oc - Q_; sh = (h + 1) & (H_ - 1); }  // y1 = roll(g0,-1,H)

    size_t src = ((((size_t)n * C_ + ic) * H_ + sh) * W_) + (size_t)t * 4;
    size_t dst = ((((size_t)n * C_ + oc) * H_ + h ) * W_) + (size_t)t * 4;

    float4v v = __builtin_nontemporal_load((const float4v*)(x + src));
    __builtin_nontemporal_store(v, (float4v*)(out + dst));
}

// -----------------------------------------------------------------------------
// Kernel B: output channels [128,256). Roll by +/-1 element along W inside each
// 256B row. Stage the source row into LDS with the CDNA5 async-to-LDS path
// (global_load_async_to_lds_b128, tracked by ASYNCcnt), wait with
// s_wait_asynccnt, then read the 4 rotated dwords with explicit ds_load_b32
// (lane stride is 4 dwords: consecutive lanes hit banks 4 apart) and do an aligned
// float4 NT store.
//
// Mapping: 256 threads = 16 rows/block, 16 threads/row. A wave32 covers
// exactly 2 rows and only reads rows it staged itself, so the per-wave
// ASYNCcnt wait is sufficient ordering -- no workgroup barrier needed.
//
// The LDS addresses are derived from &smem (ptrtoint -> low 32 bits of the
// flat address == LDS byte offset), which both matches what the async
// instruction's LDS-address VGPR expects (HW adds LDS_BASE) and makes smem
// escape so the asm "memory" clobbers legally cover it. The ds_load block
// ends with s_wait_dscnt 0x0 so the outputs are architecturally ready when
// the compiler treats them as live.
// -----------------------------------------------------------------------------
__global__ void __launch_bounds__(256)
roll_w_kernel(const float* __restrict__ x, float* __restrict__ out) {
    __shared__ float smem[16 * W_];          // 16 rows * 256B = 4KB

    int tid = threadIdx.x;
    int row = tid >> 4;                      // 0..15: row slot in LDS
    int t   = tid & 15;                      // float4 slot within row

    int rid = blockIdx.x * 16 + row;         // row id over [n:5][c2:7][h:6]
    int h   = rid & (H_ - 1);
    int c2  = (rid >> 6) & 127;              // 0..127 -> output channel 128+c2
    int n   = rid >> 13;

    int oc = 128 + c2;
    int ic, s;
    if (c2 < Q_) { ic = oc + Q_; s = -1; }   // y2 = roll(g3,+1,W): out[w]=in[w-1]
    else         { ic = oc - Q_; s = +1; }   // y3 = roll(g2,-1,W): out[w]=in[w+1]

    const float* gsrc = x + ((((size_t)n * C_ + ic) * H_ + h) * W_) + (size_t)t * 4;

    unsigned ldsBase = (unsigned)(size_t)smem;         // LDS byte offset of smem[0]
    unsigned ldsDst  = ldsBase + (unsigned)((row * W_ + t * 4) * 4);

    // Async DMA 16B/lane from global into this wave's LDS rows (ASYNCcnt++).
    asm volatile("global_load_async_to_lds_b128 %0, %1, off"
                 :: "v"(ldsDst), "v"(gsrc)
                 : "memory");
    // Wait for this wave's async copies to land in LDS.
    asm volatile("s_wait_asynccnt 0x0" ::: "memory");

    int base = row * W_;
    int w0   = t * 4;
    unsigned a0 = ldsBase + (unsigned)((base + ((w0 + 0 + s) & (W_ - 1))) * 4);
    unsigned a1 = ldsBase + (unsigned)((base + ((w0 + 1 + s) & (W_ - 1))) * 4);
    unsigned a2 = ldsBase + (unsigned)((base + ((w0 + 2 + s) & (W_ - 1))) * 4);
    unsigned a3 = ldsBase + (unsigned)((base + ((w0 + 3 + s) & (W_ - 1))) * 4);

    float e0, e1, e2, e3;
    asm volatile("ds_load_b32 %0, %4\n\t"
                 "ds_load_b32 %1, %5\n\t"
                 "ds_load_b32 %2, %6\n\t"
                 "ds_load_b32 %3, %7\n\t"
                 "s_wait_dscnt 0x0"
                 : "=&v"(e0), "=&v"(e1), "=&v"(e2), "=&v"(e3)
                 : "v"(a0), "v"(a1), "v"(a2), "v"(a3)
                 : "memory");

    float4v v;
    v.x = e0; v.y = e1; v.z = e2; v.w = e3;

    float* gdst = out + ((((size_t)n * C_ + oc) * H_ + h) * W_) + (size_t)w0;
    __builtin_nontemporal_store(v, (float4v*)gdst);
}

extern "C" void kernel_launch(void* const* d_in, const int* in_sizes, int n_in,
                              void* d_out, int out_size, void* d_ws, size_t ws_size,
                              hipStream_t stream) {
    (void)in_sizes; (void)n_in; (void)out_size; (void)d_ws; (void)ws_size;
    const float* x = (const float*)d_in[0];
    float* out = (float*)d_out;

    // Kernel A: N*128*H*W/4 float4 = 4,194,304 threads -> 16384 blocks of 256.
    roll_h_kernel<<<16384, 256, 0, stream>>>(x, out);
    // Kernel B: N*128*H = 262,144 rows, 16 rows/block -> 16384 blocks of 256.
    roll_w_kernel<<<16384, 256, 0, stream>>>(x, out);
}